// CorrFPNHead_257698038036
// MI455X (gfx1250) — compile-verified
//
#include <hip/hip_runtime.h>

#define PATCH 17
#define NDISP 289
#define CFEAT 256
#define BATCH 2

typedef __attribute__((ext_vector_type(2))) float v2f;
typedef __attribute__((ext_vector_type(8))) float v8f;

// ---------------------------------------------------------------------------
// Correlation: out[b,d,h,w] = sum_c x[b,c,h,w] * ref[b,c,h+di-8,w+dj-8]
// ~3% of total FLOPs -> VALU fmac path with x slice staged in LDS.
// ---------------------------------------------------------------------------
__global__ __launch_bounds__(256)
void corr_kernel(const float* __restrict__ x, const float* __restrict__ ref,
                 float* __restrict__ out, int H, int W)
{
    const int bh = blockIdx.x;
    const int b  = bh / H;
    const int h  = bh % H;
    const int w0 = blockIdx.y * 16;
    const int tw = threadIdx.x;     // 0..15 : spatial column
    const int ty = threadIdx.y;     // 0..15 : displacement slice
    const int w  = w0 + tw;
    const int tid = ty * 16 + tw;

    __shared__ float xs[CFEAT][16];     // 16 KB: x[b, :, h, w0..w0+15]
    for (int i = tid; i < CFEAT * 16; i += 256) {
        const int c = i >> 4, ww = i & 15;
        xs[c][ww] = x[(((size_t)b * CFEAT + c) * H + h) * W + w0 + ww];
    }
    __syncthreads();

    const size_t chanStride = (size_t)H * W;
    for (int d = ty; d < NDISP; d += 16) {
        const int di = d / PATCH, dj = d % PATCH;
        const int hh = h + di - 8;
        const int ww = w + dj - 8;
        float acc = 0.f;
        if (hh >= 0 && hh < H && ww >= 0 && ww < W) {
            const float* rp = ref + ((size_t)(b * CFEAT) * H + hh) * W + ww;
            #pragma unroll 4
            for (int c = 0; c < CFEAT; ++c)
                acc = fmaf(xs[c][tw], rp[(size_t)c * chanStride], acc);
        }
        out[(((size_t)b * NDISP + d) * H + h) * W + w] = acc;
    }
}

// ---------------------------------------------------------------------------
// Weight pre-pack: wT[((ky*3+kx)*256 + o)*CinP + ci] = w[o][ci][ky][kx] (0 pad)
// Makes each lane's A pair (ci, ci+1) one contiguous global b64 and removes
// all channel-tail guards from the WMMA loop (padding multiplies by zero).
// ---------------------------------------------------------------------------
__global__ __launch_bounds__(256)
void pack_weights(const float* __restrict__ w, float* __restrict__ wT,
                  int Cin, int CinP)
{
    const size_t total = (size_t)9 * 256 * CinP;
    size_t i = (size_t)blockIdx.x * 256 + threadIdx.x;
    if (i >= total) return;
    const int ci = (int)(i % CinP);
    size_t t = i / CinP;
    const int o   = (int)(t % 256);
    const int tap = (int)(t / 256);            // ky*3 + kx
    wT[i] = (ci < Cin) ? w[((size_t)o * Cin + ci) * 9 + tap] : 0.f;
}

// ---------------------------------------------------------------------------
// 3x3 conv as implicit GEMM on V_WMMA_F32_16X16X4_F32.
// Block = 128 threads (4 waves). Tile = 64 outch x (NT*16) columns of row h.
// band2[r][ch] channel-fastest (pad 34): B frag = single ds_load_b64.
// wT pre-packed:                         A frag = single global_load_b64.
// Interior staging via GLOBAL_LOAD_ASYNC_TO_LDS_B32 (ASYNCcnt).
// ---------------------------------------------------------------------------
template <int NT>
__global__ __launch_bounds__(128)
void conv3x3_wmma(const float* __restrict__ in0, const float* __restrict__ in1,
                  const float* __restrict__ in2, int c0, int c1, int c2,
                  const float* __restrict__ wT, int CinP,
                  const float* __restrict__ bias,
                  float* __restrict__ out, int H, int W)
{
    constexpr int ROWS = NT * 16 + 2;     // r=0 -> w0-1 ... r=NT*16+1 -> w0+NT*16
    const int Cin    = c0 + c1 + c2;
    const int wtiles = W / (16 * NT);
    int bx = blockIdx.x;
    const int wt = bx % wtiles;  bx /= wtiles;
    const int h  = bx % H;
    const int b  = bx / H;
    const int w0 = wt * (16 * NT);

    const int tid   = threadIdx.x;
    const int lane  = tid & 31;
    const int lmod  = lane & 15;
    const int lhalf = lane >> 4;
    const int oBase = blockIdx.y * 64 + (tid >> 5) * 16;
    const int o     = oBase + lmod;

    __shared__ __align__(16) float band2[ROWS][34];   // ch-fastest, pad 34

    v8f acc[NT];
    #pragma unroll
    for (int t = 0; t < NT; ++t) acc[t] = (v8f){0, 0, 0, 0, 0, 0, 0, 0};

    const int nChunks = (Cin + 31) >> 5;
    const size_t tapStride = (size_t)256 * CinP;

    for (int ky = 0; ky < 3; ++ky) {
        const int  hh    = h + ky - 1;
        const bool rowok = (hh >= 0) && (hh < H);

        for (int cc = 0; cc < nChunks; ++cc) {
            const int  cbase = cc << 5;
            const bool full  = (cbase + 32 <= Cin);

            // ---------------- stage 32 channels x (NT*16+2) columns ----------
            if (rowok && full) {
                // uniform per-chunk source select (boundaries are 32-aligned)
                const float* src; int cl, cs;
                if (cbase < c0)           { src = in0; cl = cbase;           cs = c0; }
                else if (cbase < c0 + c1) { src = in1; cl = cbase - c0;      cs = c1; }
                else                      { src = in2; cl = cbase - c0 - c1; cs = c2; }
                const float* rowp = src + (((size_t)b * cs + cl) * H + hh) * W;
                const size_t chS  = (size_t)H * W;

                // interior cols w0..w0+NT*16-1 : async b32, coalesced per row
                #pragma unroll
                for (int u = 0; u < NT * 4; ++u) {
                    const int e   = tid + u * 128;        // 0 .. NT*512-1
                    const int col = e % (NT * 16);
                    const int ch  = e / (NT * 16);
                    const float* gp = rowp + (size_t)ch * chS + (w0 + col);
                    const unsigned la = (unsigned)(size_t)&band2[col + 1][ch];
                    asm volatile("global_load_async_to_lds_b32 %0, %1, off"
                                 :: "v"(la), "v"(gp) : "memory");
                }
                // edge columns w0-1 / w0+NT*16 (zero at image borders)
                if (tid < 64) {
                    const int ch   = tid & 31;
                    const int side = tid >> 5;
                    const int gw   = side ? (w0 + NT * 16) : (w0 - 1);
                    const int r    = side ? (NT * 16 + 1) : 0;
                    float v = 0.f;
                    if (gw >= 0 && gw < W) v = rowp[(size_t)ch * chS + gw];
                    band2[r][ch] = v;
                }
                asm volatile("s_wait_asynccnt 0x0" ::: "memory");
            } else if (rowok) {
                // guarded tail-chunk staging (may cross tensor boundary / OOB)
                for (int i = tid; i < 32 * ROWS; i += 128) {
                    const int ch = i & 31;
                    const int r  = i >> 5;
                    const int ci = cbase + ch;
                    const int gw = w0 - 1 + r;
                    float v = 0.f;
                    if (ci < Cin && gw >= 0 && gw < W) {
                        if (ci < c0)
                            v = in0[(((size_t)b * c0 + ci) * H + hh) * W + gw];
                        else if (ci < c0 + c1)
                            v = in1[(((size_t)b * c1 + (ci - c0)) * H + hh) * W + gw];
                        else
                            v = in2[(((size_t)b * c2 + (ci - c0 - c1)) * H + hh) * W + gw];
                    }
                    band2[r][ch] = v;
                }
            } else {
                for (int i = tid; i < 32 * ROWS; i += 128)
                    band2[i >> 5][i & 31] = 0.f;
            }
            __syncthreads();

            // ---------------- 3 taps x 8 K-quads x NT WMMAs ------------------
            const float* aB = wT + (size_t)(ky * 3) * tapStride
                                 + (size_t)o * CinP + cbase + 2 * lhalf;
            #pragma unroll
            for (int kx = 0; kx < 3; ++kx) {
                const float* aKx = aB + (size_t)kx * tapStride;
                #pragma unroll
                for (int q = 0; q < 8; ++q) {
                    const v2f af = *(const v2f*)(aKx + q * 4);   // global b64
                    const int kk = q * 4 + 2 * lhalf;
                    #pragma unroll
                    for (int t = 0; t < NT; ++t) {
                        const v2f bf =
                            *(const v2f*)(&band2[t * 16 + lmod + kx][kk]); // ds b64
                        acc[t] = __builtin_amdgcn_wmma_f32_16x16x4_f32(
                                     false, af, false, bf, (short)0, acc[t],
                                     false, false);
                    }
                }
            }
            __syncthreads();
        }
    }

    // ---- epilogue: C/D layout M = r + 8*lhalf, N = lmod ----
    #pragma unroll
    for (int t = 0; t < NT; ++t) {
        const int wcol = w0 + t * 16 + lmod;
        #pragma unroll
        for (int r = 0; r < 8; ++r) {
            const int oc = oBase + r + 8 * lhalf;
            out[(((size_t)b * CFEAT + oc) * H + h) * W + wcol] = acc[t][r] + bias[oc];
        }
    }
}

// ---------------------------------------------------------------------------
// dst[B,256,2H,2W] += nearest-upsample(src[B,256,H,W])
// ---------------------------------------------------------------------------
__global__ __launch_bounds__(256)
void upadd_kernel(float* __restrict__ dst, const float* __restrict__ src,
                  int Hs, int Ws)
{
    const int Hd = Hs * 2, Wd = Ws * 2;
    const size_t n = (size_t)BATCH * CFEAT * Hd * Wd;
    size_t i = (size_t)blockIdx.x * blockDim.x + threadIdx.x;
    if (i >= n) return;
    const int wd = (int)(i % Wd);
    size_t t = i / Wd;
    const int hd = (int)(t % Hd);
    t /= Hd;                                  // t = b*256 + c
    dst[i] += src[(t * Hs + (hd >> 1)) * Ws + (wd >> 1)];
}

// ---------------------------------------------------------------------------
extern "C" void kernel_launch(void* const* d_in, const int* in_sizes, int n_in,
                              void* d_out, int out_size, void* d_ws, size_t ws_size,
                              hipStream_t stream)
{
    (void)in_sizes; (void)n_in; (void)out_size; (void)ws_size;

    const int Hs[4] = {128, 64, 32, 16};
    const int CIN_MAP  = 2 * CFEAT + NDISP;              // 801
    const int CINP_MAP = (CIN_MAP + 31) & ~31;           // 832
    const int CINP_OUT = CFEAT;                          // 256 (already /32)

    const float *x[4], *rx[4], *wm[4], *bm[4], *wo[4], *bo[4];
    for (int l = 0; l < 4; ++l) {
        x[l]  = (const float*)d_in[2 * l];
        rx[l] = (const float*)d_in[2 * l + 1];
        wm[l] = (const float*)d_in[8 + 4 * l + 0];
        bm[l] = (const float*)d_in[8 + 4 * l + 1];
        wo[l] = (const float*)d_in[8 + 4 * l + 2];
        bo[l] = (const float*)d_in[8 + 4 * l + 3];
    }

    // workspace: [corr scratch][map feats x4][packed-weight scratch]
    float* corr = (float*)d_ws;
    const size_t corrElems = (size_t)BATCH * NDISP * 128 * 128;
    float* maps[4];
    float* p = corr + corrElems;
    for (int l = 0; l < 4; ++l) {
        maps[l] = p;
        p += (size_t)BATCH * CFEAT * Hs[l] * Hs[l];
    }
    float* wTs = p;                                      // 9*256*832 floats max
    float* outp[4];
    float* q = (float*)d_out;
    for (int l = 0; l < 4; ++l) {
        outp[l] = q;
        q += (size_t)BATCH * CFEAT * Hs[l] * Hs[l];
    }

    const size_t mapWT = (size_t)9 * 256 * CINP_MAP;
    const size_t outWT = (size_t)9 * 256 * CINP_OUT;

    // per-level: correlation, pack map weights, 801->256 map conv (WMMA)
    for (int l = 0; l < 4; ++l) {
        const int H = Hs[l], W = Hs[l];
        corr_kernel<<<dim3(BATCH * H, W / 16), dim3(16, 16), 0, stream>>>(
            x[l], rx[l], corr, H, W);
        pack_weights<<<dim3((unsigned)((mapWT + 255) / 256)), 256, 0, stream>>>(
            wm[l], wTs, CIN_MAP, CINP_MAP);
        if (W >= 32) {
            conv3x3_wmma<2><<<dim3(BATCH * H * (W / 32), 4), 128, 0, stream>>>(
                x[l], rx[l], corr, CFEAT, CFEAT, NDISP, wTs, CINP_MAP,
                bm[l], maps[l], H, W);
        } else {
            conv3x3_wmma<1><<<dim3(BATCH * H * (W / 16), 4), 128, 0, stream>>>(
                x[l], rx[l], corr, CFEAT, CFEAT, NDISP, wTs, CINP_MAP,
                bm[l], maps[l], H, W);
        }
    }

    // top-down nearest-upsample accumulation
    for (int l = 3; l >= 1; --l) {
        const int Hsrc = Hs[l];
        const size_t n = (size_t)BATCH * CFEAT * (2 * Hsrc) * (2 * Hsrc);
        upadd_kernel<<<dim3((unsigned)((n + 255) / 256)), 256, 0, stream>>>(
            maps[l - 1], maps[l], Hsrc, Hsrc);
    }

    // 256->256 out convs (WMMA) straight into d_out
    for (int l = 0; l < 4; ++l) {
        const int H = Hs[l], W = Hs[l];
        pack_weights<<<dim3((unsigned)((outWT + 255) / 256)), 256, 0, stream>>>(
            wo[l], wTs, CFEAT, CINP_OUT);
        if (W >= 32) {
            conv3x3_wmma<2><<<dim3(BATCH * H * (W / 32), 4), 128, 0, stream>>>(
                maps[l], nullptr, nullptr, CFEAT, 0, 0, wTs, CINP_OUT,
                bo[l], outp[l], H, W);
        } else {
            conv3x3_wmma<1><<<dim3(BATCH * H * (W / 16), 4), 128, 0, stream>>>(
                maps[l], nullptr, nullptr, CFEAT, 0, 0, wTs, CINP_OUT,
                bo[l], outp[l], H, W);
        }
    }
}